// GATGraphClassifier_3_13606456394544
// MI455X (gfx1250) — compile-verified
//
#include <hip/hip_runtime.h>
#include <math.h>

// ---------------------------------------------------------------------------
// GAT graph classifier for MI455X (gfx1250, wave32).
//  * GEMMs: V_WMMA_F32_16X16X4_F32, 8 waves/block, A tile staged in LDS via
//    GLOBAL_LOAD_ASYNC_TO_LDS_B128 (ASYNCcnt) + s_wait_asynccnt.
//  * Edge softmax: 2 passes (atomicMax, unnormalized exp-accumulate) +
//    normalize; gathers/atomics land in the 192MB L2 (node features = 25.6MB).
//  * Edge accumulate: one lane per channel -> coalesced 64B gathers, one
//    global_atomic_add_f32 per lane.
// ---------------------------------------------------------------------------

typedef float v2f __attribute__((ext_vector_type(2)));
typedef float v8f __attribute__((ext_vector_type(8)));
typedef int   v4i __attribute__((ext_vector_type(4)));

#define AS1 __attribute__((address_space(1)))
#define AS3 __attribute__((address_space(3)))

#if defined(__HIP_DEVICE_COMPILE__) &&                                   \
    __has_builtin(__builtin_amdgcn_global_load_async_to_lds_b128) &&     \
    __has_builtin(__builtin_amdgcn_s_wait_asynccnt)
#define HAVE_ASYNC_LDS 1
#else
#define HAVE_ASYNC_LDS 0
#endif

// C[M,Nc] = A[M,K] * B[K,Nc], row-major f32.  Requires K == 128, M%16 == 0,
// Nc%16 == 0.  Launch: grid(M/16), block(32, Nc/16) -> one wave per 16x16
// C tile, all waves share the A tile staged in LDS.
__global__ void gemm16_wmma(const float* __restrict__ A,
                            const float* __restrict__ B,
                            float* __restrict__ C,
                            int M, int K, int Nc) {
  // Row pad 132 floats: 528B row stride -> 16B-aligned b128 stores, and the
  // 16 A-fragment lanes hit banks (r*4 + k) % 64, all distinct.
  __shared__ float Alds[16][132];

  const int m0   = blockIdx.x * 16;
  const int lane = threadIdx.x;          // 0..31
  const int wave = threadIdx.y;          // 0..Nc/16-1
  const int n0   = wave * 16;
  const int nthr = blockDim.x * blockDim.y;
  const int tid  = wave * 32 + lane;

  // ---- Stage A tile [16][K] into LDS (float4 / b128 granularity) ----
  const int vec_total = 16 * K / 4;      // number of float4 chunks
  for (int v = tid; v < vec_total; v += nthr) {
    const int rowA = (v * 4) / K;
    const int colA = (v * 4) % K;
    const float* gp = A + (size_t)(m0 + rowA) * K + colA;
    float*       lp = &Alds[rowA][colA];
#if HAVE_ASYNC_LDS
    __builtin_amdgcn_global_load_async_to_lds_b128((AS1 v4i*)gp, (AS3 v4i*)lp,
                                                   0, 0);
#else
    *(float4*)lp = *(const float4*)gp;
#endif
  }
#if HAVE_ASYNC_LDS
  __builtin_amdgcn_s_wait_asynccnt(0);
#endif
  __syncthreads();

  const int half = lane >> 4;            // 0: K 0/1, 1: K 2/3 (ISA 7.12.2)
  const int r    = lane & 15;
  const float* Brow = B + (size_t)(2 * half) * Nc + (n0 + r);
  __builtin_prefetch(Brow, 0, 3);        // global_prefetch_b8 (B reused chipwide)

  v8f c = {};
  for (int k0 = 0; k0 < K; k0 += 4) {
    v2f a, b;
    a.x = Alds[r][k0 + 2 * half];
    a.y = Alds[r][k0 + 2 * half + 1];
    b.x = Brow[(size_t)(k0)     * Nc];
    b.y = Brow[(size_t)(k0 + 1) * Nc];
    // 8 args: (neg_a, A, neg_b, B, c_mod, C, reuse_a, reuse_b)
    c = __builtin_amdgcn_wmma_f32_16x16x4_f32(false, a, false, b,
                                              (short)0, c, false, false);
  }
#pragma unroll
  for (int j = 0; j < 8; ++j) {
    const int row = m0 + j + 8 * half;   // C/D layout: VGPR j -> M=j / M=8+j
    C[(size_t)row * Nc + n0 + r] = c[j];
  }
}

// Per (node, head): attention scalars; also init m = -inf, s = 0.
__global__ void att_prep(const float* __restrict__ xl,
                         const float* __restrict__ asrc,
                         const float* __restrict__ adst,
                         float* __restrict__ a_s, float* __restrict__ a_d,
                         float* __restrict__ m, float* __restrict__ s,
                         int N, int H, int C) {
  const int tid = blockIdx.x * blockDim.x + threadIdx.x;
  if (tid >= N * H) return;
  const int h = tid % H;
  const float* v = xl + (size_t)tid * C;   // xl[n,h,c] flat: (n*H+h)*C + c
  float ds = 0.f, dd = 0.f;
  for (int c = 0; c < C; ++c) {
    const float xv = v[c];
    ds += xv * asrc[h * C + c];
    dd += xv * adst[h * C + c];
  }
  a_s[tid] = ds;
  a_d[tid] = dd;
  m[tid]   = -1e30f;
  s[tid]   = 0.f;
}

__device__ __forceinline__ void atomicMaxFloat(float* addr, float val) {
  if (val >= 0.f) atomicMax((int*)addr, __float_as_int(val));
  else            atomicMin((unsigned int*)addr, __float_as_uint(val));
}

// Pass 1: segment max of leaky_relu(a_s[src]+a_d[dst]) over dst.
__global__ void edge_max(const long long* __restrict__ eidx, int E, int N,
                         const float* __restrict__ a_s,
                         const float* __restrict__ a_d,
                         float* __restrict__ m, int H) {
  const int t  = blockIdx.x * blockDim.x + threadIdx.x;
  const int ne = E + N;
  if (t >= ne) return;
  int src, dst;
  if (t < E) { src = (int)eidx[t]; dst = (int)eidx[(size_t)E + t]; }
  else       { src = dst = t - E; }                 // self loops
  for (int hh = 0; hh < H; ++hh) {
    float v = a_s[src * H + hh] + a_d[dst * H + hh];
    v = v > 0.f ? v : 0.2f * v;                     // leaky_relu 0.2
    atomicMaxFloat(&m[dst * H + hh], v);
  }
}

// Pass 2: one lane per (edge, head, channel).  C must be 16.
// p = exp(e - m[dst]); s[dst] += p (lane c==0); acc[dst,h,c] += p*xl[src,h,c].
__global__ void edge_accum(const long long* __restrict__ eidx, int E, int N,
                           const float* __restrict__ a_s,
                           const float* __restrict__ a_d,
                           const float* __restrict__ m,
                           float* __restrict__ s,
                           const float* __restrict__ xl,
                           float* __restrict__ acc, int H) {
  const long long t = (long long)blockIdx.x * blockDim.x + threadIdx.x;
  const int ne = E + N;
  if (t >= (long long)ne * H * 16) return;
  const int c  = (int)(t & 15);
  const int hc = (int)(t >> 4);
  const int hh = hc % H;
  const int e  = hc / H;
  int src, dst;
  if (e < E) { src = (int)eidx[e]; dst = (int)eidx[(size_t)E + e]; }
  else       { src = dst = e - E; }
  float v = a_s[src * H + hh] + a_d[dst * H + hh];
  v = v > 0.f ? v : 0.2f * v;
  const float p = expf(v - m[dst * H + hh]);
  if (c == 0) atomicAdd(&s[dst * H + hh], p);
  atomicAdd(&acc[((size_t)dst * H + hh) * 16 + c],
            p * xl[((size_t)src * H + hh) * 16 + c]);
}

// out = act(acc / s + bias);  act: 0 = ELU, 1 = ReLU
__global__ void finalize(const float* __restrict__ acc,
                         const float* __restrict__ s,
                         const float* __restrict__ bias,
                         float* __restrict__ out,
                         int N, int H, int C, int act) {
  const int tid = blockIdx.x * blockDim.x + threadIdx.x;
  if (tid >= N * H * C) return;
  const int n   = tid / (H * C);
  const int rem = tid % (H * C);
  const int h   = rem / C;
  float v = acc[tid] / s[n * H + h] + bias[rem];
  if (act == 0) v = v > 0.f ? v : (expf(v) - 1.f);   // ELU
  else          v = v > 0.f ? v : 0.f;               // ReLU
  out[tid] = v;
}

__global__ void pool_kernel(const float* __restrict__ h3,
                            const long long* __restrict__ batch,
                            float* __restrict__ sums, float* __restrict__ cnt,
                            int N, int C) {
  const int t = blockIdx.x * blockDim.x + threadIdx.x;
  if (t >= N * C) return;
  const int n = t / C, c = t % C;
  const int g = (int)batch[n];
  atomicAdd(&sums[g * C + c], h3[t]);
  if (c == 0) atomicAdd(&cnt[g], 1.0f);
}

__global__ void final_linear(const float* __restrict__ sums,
                             const float* __restrict__ cnt,
                             const float* __restrict__ Wl,
                             const float* __restrict__ bl,
                             float* __restrict__ out, int G, int C, int O) {
  const int t = blockIdx.x * blockDim.x + threadIdx.x;
  if (t >= G * O) return;
  const int g = t / O, o = t % O;
  const float inv = 1.0f / fmaxf(cnt[g], 1.0f);
  float a = bl[o];
  for (int c = 0; c < C; ++c) a += (sums[g * C + c] * inv) * Wl[c * O + o];
  out[t] = a;
}

// ---------------------------------------------------------------------------

static void run_gat_layer(const float* X, const long long* eidx, int E, int N,
                          const float* W, const float* asrc, const float* adst,
                          const float* bias, int K, int H, int C, int act,
                          float* xl, float* acc, float* a_s, float* a_d,
                          float* m, float* s, float* out, hipStream_t stream) {
  const int D = H * C;
  dim3 blk(32, D / 16);
  gemm16_wmma<<<dim3(N / 16), blk, 0, stream>>>(X, W, xl, N, K, D);
  (void)hipMemsetAsync(acc, 0, (size_t)N * D * sizeof(float), stream);
  const int nh = N * H;
  att_prep<<<(nh + 255) / 256, 256, 0, stream>>>(xl, asrc, adst, a_s, a_d, m, s,
                                                 N, H, C);
  const int ne = E + N;
  edge_max<<<(ne + 255) / 256, 256, 0, stream>>>(eidx, E, N, a_s, a_d, m, H);
  const long long tot = (long long)ne * H * 16;
  edge_accum<<<(unsigned)((tot + 255) / 256), 256, 0, stream>>>(eidx, E, N, a_s,
                                                                a_d, m, s, xl,
                                                                acc, H);
  const int nd = N * D;
  finalize<<<(nd + 255) / 256, 256, 0, stream>>>(acc, s, bias, out, N, H, C, act);
}

extern "C" void kernel_launch(void* const* d_in, const int* in_sizes, int n_in,
                              void* d_out, int out_size, void* d_ws, size_t ws_size,
                              hipStream_t stream) {
  const float*     x     = (const float*)d_in[0];
  const long long* eidx  = (const long long*)d_in[1];
  const long long* batch = (const long long*)d_in[2];
  const float* W1  = (const float*)d_in[3];
  const float* as1 = (const float*)d_in[4];
  const float* ad1 = (const float*)d_in[5];
  const float* b1  = (const float*)d_in[6];
  const float* W2  = (const float*)d_in[7];
  const float* as2 = (const float*)d_in[8];
  const float* ad2 = (const float*)d_in[9];
  const float* b2  = (const float*)d_in[10];
  const float* W3  = (const float*)d_in[11];
  const float* as3 = (const float*)d_in[12];
  const float* ad3 = (const float*)d_in[13];
  const float* b3  = (const float*)d_in[14];
  const float* Wl  = (const float*)d_in[15];
  const float* bl  = (const float*)d_in[16];

  const int N = in_sizes[0] / 128;   // 50000
  const int E = in_sizes[1] / 2;     // 800000
  const int H = 8, C = 16, D = 128;
  const int OUT = 10;
  const int G = out_size / OUT;      // 64

  // Workspace carve-up (f32)
  float* xl     = (float*)d_ws;
  float* acc    = xl  + (size_t)N * D;
  float* h      = acc + (size_t)N * D;
  float* a_s    = h   + (size_t)N * D;
  float* a_d    = a_s + (size_t)N * H;
  float* m      = a_d + (size_t)N * H;
  float* s      = m   + (size_t)N * H;
  float* h3     = s   + (size_t)N * H;
  float* pooled = h3  + (size_t)N * C;
  float* cnt    = pooled + (size_t)G * C;
  (void)ws_size; (void)n_in;

  // Layer 1: x[N,128] -> h[N,128], ELU
  run_gat_layer(x, eidx, E, N, W1, as1, ad1, b1, 128, H, C, /*elu*/0,
                xl, acc, a_s, a_d, m, s, h, stream);
  // Layer 2: h -> h (safe: gemm output goes to xl), ELU
  run_gat_layer(h, eidx, E, N, W2, as2, ad2, b2, 128, H, C, /*elu*/0,
                xl, acc, a_s, a_d, m, s, h, stream);
  // Layer 3: h[N,128] -> h3[N,16], 1 head, ReLU
  run_gat_layer(h, eidx, E, N, W3, as3, ad3, b3, 128, 1, C, /*relu*/1,
                xl, acc, a_s, a_d, m, s, h3, stream);

  // Global mean pool + final linear
  (void)hipMemsetAsync(pooled, 0, (size_t)(G * C + G) * sizeof(float), stream);
  const int nc = N * C;
  pool_kernel<<<(nc + 255) / 256, 256, 0, stream>>>(h3, batch, pooled, cnt, N, C);
  const int go = G * OUT;
  final_linear<<<(go + 255) / 256, 256, 0, stream>>>(pooled, cnt, Wl, bl,
                                                     (float*)d_out, G, C, OUT);
}